// FractionalDelayLines_81054622810655
// MI455X (gfx1250) — compile-verified
//
#include <hip/hip_runtime.h>
#include <hip/hip_bf16.h>
#include <math.h>

// Problem constants (from reference)
#define N_LINES    128
#define BUF_LEN    48000
#define N_BINS     64
#define KFFT       (2 * BUF_LEN)              // 96000
#define T0         (BUF_LEN - N_BINS - 1)     // 47935: first tap index
#define HALVES     4
#define S_CHUNK    (BUF_LEN / HALVES)         // 12000 s-values per WG
#define H_COUNT    (S_CHUNK + N_BINS - 1)     // 12063 h-table entries per WG
#define THREADS    1024                       // 32 wave32 waves

typedef float v2f __attribute__((ext_vector_type(2)));
typedef float v8f __attribute__((ext_vector_type(8)));

// out[n] = sum_s buf[n,s] * g_n(s),  g_n(s) = sum_j rf[n,j] * h_n[T0+j-s]
// h_n[m] = -(-1)^m * (sin(pi d)/K) * cot(pi (m-d)/K)
__global__ __launch_bounds__(THREADS)
void fdl_main(const float* __restrict__ buffer,
              const float* __restrict__ inputs,
              const float* __restrict__ delays,
              const float* __restrict__ rfilt,
              float* __restrict__ partials)
{
    __shared__ float Hs[H_COUNT + 1];
    __shared__ float red[32];

    const int wg    = blockIdx.x;
    const int n     = wg >> 2;           // delay line
    const int half  = wg & 3;            // quarter of the s-range
    const int sBase = half * S_CHUNK;
    const int mLow  = T0 - (sBase + S_CHUNK - 1);   // smallest m this WG needs

    const int tid  = threadIdx.x;
    const int lane = tid & 31;
    const int wave = tid >> 5;

    // Per-line constants (redundant per thread, trivially cheap)
    const float d       = delays[n];                 // SCALE == 1.0
    const float invK    = 1.0f / (float)KFFT;
    const float piOverK = (float)M_PI * invK;
    float Sd, Cd;
    __sincosf(d * piOverK, &Sd, &Cd);                // sin/cos(pi d / K)
    const float S = sinpif(d) * invK;                // sin(pi d)/K (stable near integer d)

    // ---- Phase A: build h-table slice in LDS ----
    // cos/sin(pi(m-d)/K) via angle addition: never form the angle at large m in fp32.
    for (int li = tid; li < H_COUNT; li += THREADS) {
        const int m = mLow + li;
        float sm, cm;
        __sincosf((float)m * piOverK, &sm, &cm);
        const float num = cm * Cd + sm * Sd;         // cos(pi(m-d)/K)
        const float den = sm * Cd - cm * Sd;         // sin(pi(m-d)/K)
        const float sgn = (m & 1) ? 1.0f : -1.0f;    // = -(-1)^m
        Hs[li] = sgn * S * num / den;
    }
    __syncthreads();

    // ---- Phase B: WMMA Toeplitz correlation ----
    // Step p of 16 contracts j = 4p..4p+3:
    //   A[r][k] = Hs[(T0 + 4p + k - s_r) - mLow],  B[k][c] = rf[4p+k] (all columns)
    // A layout (32-bit 16x4): lanes 0-15 hold K=0,1 ; lanes 16-31 hold K=2,3.
    const int kb  = (lane >> 4) << 1;                // 0 or 2
    const int row = lane & 15;                       // A row handled by this lane pair (its D column)
    const float* rf = rfilt + n * N_BINS;
    v2f barr[16];
    #pragma unroll
    for (int p = 0; p < 16; ++p) {
        barr[p].x = rf[4 * p + kb + 0];
        barr[p].y = rf[4 * p + kb + 1];
    }

    const float* bufrow  = buffer + (size_t)n * BUF_LEN;
    const float  inp     = inputs[n];
    const int    rowbase = (lane < 16) ? 0 : 8;      // D rows this lane holds in VGPRs

    float partial = 0.0f;
    for (int b = wave; b < S_CHUNK / 16; b += 32) {  // uniform per wave -> EXEC all-1 for WMMA
        const int sl0 = b * 16;                      // local s base of this 16-s block
        __builtin_prefetch(bufrow + sBase + sl0 + 1 + 1024, 0, 0);  // global_prefetch_b8

        // li = 4p + kb + (S_CHUNK-1) - (sl0 + row)
        const int abase = (S_CHUNK - 1) - sl0 - row + kb;
        v8f acc = {0.f, 0.f, 0.f, 0.f, 0.f, 0.f, 0.f, 0.f};
        #pragma unroll
        for (int p = 0; p < 16; ++p) {
            const int li = abase + 4 * p;
            v2f a;
            a.x = Hs[li];
            a.y = Hs[li + 1];
            acc = __builtin_amdgcn_wmma_f32_16x16x4_f32(
                false, a, false, barr[p], (short)0, acc, false, false);
        }

        // D[r][c] = g(sBase+sl0+r) for every column c (16x redundant; /16 at the end).
        #pragma unroll
        for (int j = 0; j < 8; ++j) {
            const int s  = sBase + sl0 + rowbase + j;
            const float bv = (s == BUF_LEN - 1) ? inp : bufrow[s + 1];  // rolled buffer
            partial += acc[j] * bv;
        }
    }

    // ---- reduction: wave shuffle tree, then cross-wave via LDS ----
    for (int off = 16; off > 0; off >>= 1)
        partial += __shfl_xor(partial, off, 32);
    if (lane == 0) red[wave] = partial;
    __syncthreads();
    if (wave == 0) {
        float v = red[lane];
        for (int off = 16; off > 0; off >>= 1)
            v += __shfl_xor(v, off, 32);
        if (lane == 0) partials[wg] = v * (1.0f / 16.0f);   // undo column redundancy
    }
}

__global__ void fdl_combine(const float* __restrict__ partials,
                            float* __restrict__ out)
{
    const int n = threadIdx.x;
    if (n < N_LINES) {
        out[n] = (partials[4 * n + 0] + partials[4 * n + 1])
               + (partials[4 * n + 2] + partials[4 * n + 3]);
    }
}

extern "C" void kernel_launch(void* const* d_in, const int* in_sizes, int n_in,
                              void* d_out, int out_size, void* d_ws, size_t ws_size,
                              hipStream_t stream) {
    const float* buffer = (const float*)d_in[0];   // (128, 48000) f32
    const float* inputs = (const float*)d_in[1];   // (128,)       f32
    const float* delays = (const float*)d_in[2];   // (128,)       f32
    const float* rfilt  = (const float*)d_in[3];   // (128, 64)    f32
    float* out = (float*)d_out;                    // (128, 1)     f32
    float* ws  = (float*)d_ws;                     // 512 floats of scratch

    fdl_main<<<dim3(N_LINES * HALVES), dim3(THREADS), 0, stream>>>(
        buffer, inputs, delays, rfilt, ws);
    fdl_combine<<<dim3(1), dim3(N_LINES), 0, stream>>>(ws, out);
}